// GCN_29283087024285
// MI455X (gfx1250) — compile-verified
//
#include <hip/hip_runtime.h>
#include <hip/hip_bf16.h>

typedef float v2f __attribute__((ext_vector_type(2)));
typedef float v8f __attribute__((ext_vector_type(8)));

#define N_NODES 100000
#define N_EDGES 3200000
#define D_IN    512
#define D_H     64
#define D_OUT   40
#define D_PAD   48   // 40 padded to 3 WMMA N-tiles

// ---------------------------------------------------------------------------
// GEMM1: h1[N,64] = x[N,512] @ W1[512,64]  via V_WMMA_F32_16X16X4_F32
// Block = 256 threads (8 waves). Each wave computes a 16x64 tile (4 N-tiles).
// W1 staged through LDS in 128-row K-chunks, K-pair interleaved so every
// B fragment is a single aligned ds_load_b64.
// ---------------------------------------------------------------------------
__global__ __launch_bounds__(256) void gcn_gemm1(const float* __restrict__ x,
                                                 const float* __restrict__ W1,
                                                 float* __restrict__ h1) {
    // bs2[p*64 + col] = { W1chunk[2p][col], W1chunk[2p+1][col] }, p in [0,64)
    __shared__ v2f bs2[64 * D_H];                 // 32 KB chunk of W1 (paired)

    const int lane = threadIdx.x & 31;
    const int wave = threadIdx.x >> 5;
    const int half = lane >> 4;                   // 0: K pair {0,1}; 1: K pair {2,3}
    const int l15  = lane & 15;

    // A-fragment row for this lane (clamped; tail stores are guarded below)
    int row  = blockIdx.x * 128 + wave * 16 + l15;
    int rowc = row < N_NODES ? row : (N_NODES - 1);
    const float* xrow = x + (size_t)rowc * D_IN;

    v8f acc0 = {0.f,0.f,0.f,0.f,0.f,0.f,0.f,0.f};
    v8f acc1 = acc0, acc2 = acc0, acc3 = acc0;

    for (int kc = 0; kc < D_IN / 128; ++kc) {
        // cooperative load of W1 chunk into LDS, interleaving K pairs
        const float* wsrc = W1 + kc * 128 * D_H;
        #pragma unroll
        for (int i = threadIdx.x; i < 64 * D_H; i += 256) {
            const int p = i >> 6;                 // pair index
            const int j = i & 63;                 // column
            v2f t;
            t.x = wsrc[(2 * p)     * D_H + j];
            t.y = wsrc[(2 * p + 1) * D_H + j];
            bs2[i] = t;
        }
        // prefetch next A chunk while we sync + compute this one
        if (kc + 1 < D_IN / 128)
            __builtin_prefetch(xrow + (kc + 1) * 128, 0, 0);
        __syncthreads();

        #pragma unroll 4
        for (int k4 = 0; k4 < 32; ++k4) {
            // A frag: 2 consecutive f32 per lane (global_load_b64)
            v2f a = *(const v2f*)(xrow + kc * 128 + k4 * 4 + half * 2);
            // B frags: one aligned ds_load_b64 per N-tile
            const int p = k4 * 2 + half;
            const v2f* brow = bs2 + p * D_H + l15;
            v2f b0 = brow[ 0];
            v2f b1 = brow[16];
            v2f b2 = brow[32];
            v2f b3 = brow[48];
            acc0 = __builtin_amdgcn_wmma_f32_16x16x4_f32(false, a, false, b0, (short)0, acc0, false, false);
            acc1 = __builtin_amdgcn_wmma_f32_16x16x4_f32(false, a, false, b1, (short)0, acc1, false, false);
            acc2 = __builtin_amdgcn_wmma_f32_16x16x4_f32(false, a, false, b2, (short)0, acc2, false, false);
            acc3 = __builtin_amdgcn_wmma_f32_16x16x4_f32(false, a, false, b3, (short)0, acc3, false, false);
        }
        __syncthreads();
    }

    // C/D layout: VGPR v holds M = v + 8*half, N = l15 (per 16x16 f32 table)
    const int rbase = blockIdx.x * 128 + wave * 16 + half * 8;
    #pragma unroll
    for (int v = 0; v < 8; ++v) {
        const int r = rbase + v;
        if (r < N_NODES) {
            float* o = h1 + (size_t)r * D_H + l15;
            o[ 0] = acc0[v];
            o[16] = acc1[v];
            o[32] = acc2[v];
            o[48] = acc3[v];
        }
    }
}

// ---------------------------------------------------------------------------
// SpMM D=64: one wave per edge; lane covers features {2l, 2l+1}.
// g1 must be zeroed first; scatter via L2 f32 atomics (h/out are L2-resident).
// ---------------------------------------------------------------------------
__global__ __launch_bounds__(256) void gcn_spmm64(const int* __restrict__ row,
                                                  const int* __restrict__ col,
                                                  const float* __restrict__ val,
                                                  const float* __restrict__ h,
                                                  float* __restrict__ out) {
    const int wid  = (int)((blockIdx.x * blockDim.x + threadIdx.x) >> 5);
    const int lane = threadIdx.x & 31;
    if (wid >= N_EDGES) return;
    const int   r = row[wid];
    const int   c = col[wid];
    const float v = val[wid];
    v2f hv = *(const v2f*)(h + (size_t)c * D_H + lane * 2);
    float* o = out + (size_t)r * D_H + lane * 2;
    atomicAdd(o,     v * hv.x);
    atomicAdd(o + 1, v * hv.y);
}

// ---------------------------------------------------------------------------
// GEMM2: h2[N,40] = relu(g1)[N,64] @ W2[64,40], N-dim padded 40->48.
// ReLU fused into the A-fragment load. W2 zero-padded + K-pair interleaved
// in LDS. 3 WMMA N-tiles, K loop of 16.
// ---------------------------------------------------------------------------
__global__ __launch_bounds__(256) void gcn_gemm2(const float* __restrict__ g1,
                                                 const float* __restrict__ W2,
                                                 float* __restrict__ h2) {
    // bs2[p*48 + j] = { W2pad[2p][j], W2pad[2p+1][j] }, p in [0,32)
    __shared__ v2f bs2[32 * D_PAD];               // 12 KB

    #pragma unroll
    for (int i = threadIdx.x; i < 32 * D_PAD; i += 256) {
        const int p = i / D_PAD, j = i - p * D_PAD;
        v2f t;
        t.x = (j < D_OUT) ? W2[(2 * p)     * D_OUT + j] : 0.f;
        t.y = (j < D_OUT) ? W2[(2 * p + 1) * D_OUT + j] : 0.f;
        bs2[i] = t;
    }
    __syncthreads();

    const int lane = threadIdx.x & 31;
    const int wave = threadIdx.x >> 5;
    const int half = lane >> 4;
    const int l15  = lane & 15;

    int row  = blockIdx.x * 128 + wave * 16 + l15;
    int rowc = row < N_NODES ? row : (N_NODES - 1);
    const float* grow = g1 + (size_t)rowc * D_H;

    v8f acc0 = {0.f,0.f,0.f,0.f,0.f,0.f,0.f,0.f};
    v8f acc1 = acc0, acc2 = acc0;

    #pragma unroll 4
    for (int k4 = 0; k4 < 16; ++k4) {
        v2f a = *(const v2f*)(grow + k4 * 4 + half * 2);
        a.x = fmaxf(a.x, 0.f);                    // fused ReLU
        a.y = fmaxf(a.y, 0.f);
        const int p = k4 * 2 + half;
        const v2f* brow = bs2 + p * D_PAD + l15;
        v2f b0 = brow[ 0];
        v2f b1 = brow[16];
        v2f b2 = brow[32];
        acc0 = __builtin_amdgcn_wmma_f32_16x16x4_f32(false, a, false, b0, (short)0, acc0, false, false);
        acc1 = __builtin_amdgcn_wmma_f32_16x16x4_f32(false, a, false, b1, (short)0, acc1, false, false);
        acc2 = __builtin_amdgcn_wmma_f32_16x16x4_f32(false, a, false, b2, (short)0, acc2, false, false);
    }

    const int rbase = blockIdx.x * 128 + wave * 16 + half * 8;
    #pragma unroll
    for (int v = 0; v < 8; ++v) {
        const int r = rbase + v;
        if (r < N_NODES) {
            float* o = h2 + (size_t)r * D_OUT + l15;
            o[ 0] = acc0[v];
            o[16] = acc1[v];
            if (l15 < 8) o[32] = acc2[v];         // cols 32..39 only
        }
    }
}

// ---------------------------------------------------------------------------
// SpMM D=40: one wave per edge; lanes cover f in [0,32), lanes 0..7 also [32,40).
// ---------------------------------------------------------------------------
__global__ __launch_bounds__(256) void gcn_spmm40(const int* __restrict__ row,
                                                  const int* __restrict__ col,
                                                  const float* __restrict__ val,
                                                  const float* __restrict__ h,
                                                  float* __restrict__ out) {
    const int wid  = (int)((blockIdx.x * blockDim.x + threadIdx.x) >> 5);
    const int lane = threadIdx.x & 31;
    if (wid >= N_EDGES) return;
    const int   r = row[wid];
    const int   c = col[wid];
    const float v = val[wid];
    const float* hrow = h + (size_t)c * D_OUT;
    float*       orow = out + (size_t)r * D_OUT;
    atomicAdd(orow + lane, v * hrow[lane]);
    if (lane < 8)
        atomicAdd(orow + 32 + lane, v * hrow[32 + lane]);
}

// ---------------------------------------------------------------------------
extern "C" void kernel_launch(void* const* d_in, const int* in_sizes, int n_in,
                              void* d_out, int out_size, void* d_ws, size_t ws_size,
                              hipStream_t stream) {
    const float* x   = (const float*)d_in[0];
    const float* W1  = (const float*)d_in[1];
    const float* W2  = (const float*)d_in[2];
    const float* val = (const float*)d_in[3];
    const int*   row = (const int*)d_in[4];
    const int*   col = (const int*)d_in[5];
    float*       out = (float*)d_out;

    float* h1 = (float*)d_ws;                       // [N,64]
    float* g1 = h1 + (size_t)N_NODES * D_H;         // [N,64]
    float* h2 = g1 + (size_t)N_NODES * D_H;         // [N,40]

    hipMemsetAsync(g1,  0, (size_t)N_NODES * D_H  * sizeof(float), stream);
    hipMemsetAsync(out, 0, (size_t)N_NODES * D_OUT * sizeof(float), stream);

    const dim3 blk(256);
    const int gemm_grid = (N_NODES + 127) / 128;
    const int spmm_grid = (int)(((long long)N_EDGES * 32 + 255) / 256);

    gcn_gemm1 <<<gemm_grid, blk, 0, stream>>>(x, W1, h1);
    gcn_spmm64<<<spmm_grid, blk, 0, stream>>>(row, col, val, h1, g1);
    gcn_gemm2 <<<gemm_grid, blk, 0, stream>>>(g1, W2, h2);
    gcn_spmm40<<<spmm_grid, blk, 0, stream>>>(row, col, val, h2, out);
}